// MultiHeadAttention_90220083019788
// MI455X (gfx1250) — compile-verified
//
#include <hip/hip_runtime.h>
#include <math.h>

// ---------------------------------------------------------------------------
// MHA: B=2, S=512, DIM=256, H=8, hd=32, fp32 throughout.
// time_k/time_v (d_in[1], d_in[2]) are dead inputs -> never read.
// Kernel 1: qkv = hidden @ {Wq,Wk,Wv}^T + bias  (V_WMMA_F32_16X16X4_F32)
// Kernel 2: per (b,h,q-tile16): scores WMMA -> masked softmax (LDS) -> ctx WMMA
// ---------------------------------------------------------------------------

typedef __attribute__((ext_vector_type(2))) float v2f;
typedef __attribute__((ext_vector_type(8))) float v8f;

#define DIMC 256
#define SEQ  512
#define NH   8
#define HD   32
#define BB   2
#define MROWS (BB * SEQ)      // 1024

__device__ __forceinline__ v8f wmma_f32(v2f a, v2f b, v8f c) {
    // D = A(16x4) * B(4x16) + C(16x16), fp32
    return __builtin_amdgcn_wmma_f32_16x16x4_f32(
        false, a, false, b, (short)0, c, false, false);
}

// ---------------------------------------------------------------------------
// QKV projection: y[m][n] = sum_k X[m][k] * W[n][k] + bias[n]
// 3 matrices x (1024/16) m-tiles x (256/16) n-tiles = 3072 tiles, 1 wave each.
// ---------------------------------------------------------------------------
__global__ void __launch_bounds__(256)
mha_qkv_proj_kernel(const float* __restrict__ hidden,
                    const float* __restrict__ Wq, const float* __restrict__ bq,
                    const float* __restrict__ Wk, const float* __restrict__ bk,
                    const float* __restrict__ Wv, const float* __restrict__ bv,
                    float* __restrict__ qkv /* [3][1024][256] */)
{
    const int waveId = (blockIdx.x * blockDim.x + threadIdx.x) >> 5;
    if (waveId >= 3 * (MROWS / 16) * (DIMC / 16)) return;   // wave-uniform

    const int mat   = waveId >> 10;          // /1024 tiles per matrix
    const int rem   = waveId & 1023;
    const int mbase = (rem >> 4) << 4;       // m-tile * 16
    const int nbase = (rem & 15) << 4;       // n-tile * 16

    const float* __restrict__ W    = (mat == 0) ? Wq : (mat == 1) ? Wk : Wv;
    const float* __restrict__ bias = (mat == 0) ? bq : (mat == 1) ? bk : bv;

    const int lane = threadIdx.x & 31;
    const int half = lane >> 4;              // 0/1
    const int l16  = lane & 15;

    const float* __restrict__ Arow = hidden + (size_t)(mbase + l16) * DIMC;
    const float* __restrict__ Brow = W      + (size_t)(nbase + l16) * DIMC;

    v8f acc = {};
#pragma unroll 8
    for (int kk = 0; kk < DIMC / 4; ++kk) {
        const int k0 = kk * 4 + 2 * half;
        v2f a = { Arow[k0], Arow[k0 + 1] };
        v2f b = { Brow[k0], Brow[k0 + 1] };
        acc = wmma_f32(a, b, acc);
    }

    const float bv_ = bias[nbase + l16];
    float* __restrict__ outp = qkv + (size_t)mat * MROWS * DIMC;
#pragma unroll
    for (int r = 0; r < 8; ++r) {
        const int m = mbase + 8 * half + r;
        outp[(size_t)m * DIMC + nbase + l16] = acc[r] + bv_;
    }
}

// ---------------------------------------------------------------------------
// Fused attention: one wave (32 threads) per (b, h, 16-query-row tile).
// grid = B*H*(S/16) = 512 blocks of 32 threads.
// ---------------------------------------------------------------------------
#define PROW (SEQ + 4)   // LDS row pad: stride 516 floats -> conflict-free WMMA reads

__global__ void __launch_bounds__(32)
mha_attn_kernel(const float* __restrict__ qkv,
                const int*   __restrict__ attn_mask,  // [B][S]
                float*       __restrict__ out)        // [B][S][DIM]
{
    __shared__ float probs[16][PROW];                 // 33 KB

    const int tile = blockIdx.x;                      // 0..511
    const int b    = tile >> 8;                       // / (H * S/16)
    const int rem  = tile & 255;
    const int h    = rem >> 5;
    const int qt   = rem & 31;
    const int qbase = qt << 4;

    const int lane = threadIdx.x;
    const int half = lane >> 4;
    const int l16  = lane & 15;

    const size_t headOff = (size_t)b * SEQ * DIMC + (size_t)h * HD;
    const float* __restrict__ Q = qkv + 0 * (size_t)MROWS * DIMC + headOff;
    const float* __restrict__ K = qkv + 1 * (size_t)MROWS * DIMC + headOff;
    const float* __restrict__ V = qkv + 2 * (size_t)MROWS * DIMC + headOff;

    // Preload the Q A-fragments for all 8 K-chunks (hd=32 = 8 * 4)
    v2f qfrag[8];
    {
        const float* __restrict__ Qrow = Q + (size_t)(qbase + l16) * DIMC;
#pragma unroll
        for (int kk = 0; kk < 8; ++kk) {
            const int k0 = kk * 4 + 2 * half;
            qfrag[kk] = (v2f){ Qrow[k0], Qrow[k0 + 1] };
        }
    }

    const float scale = 0.17677669529663687f;         // 1/sqrt(32)

    // ---- scores = (Q K^T)*scale + mask -> LDS ----
    for (int nt = 0; nt < SEQ / 16; ++nt) {
        const int nb = nt << 4;
        const float* __restrict__ Krow = K + (size_t)(nb + l16) * DIMC;
        v8f acc = {};
#pragma unroll
        for (int kk = 0; kk < 8; ++kk) {
            const int k0 = kk * 4 + 2 * half;
            v2f bf = (v2f){ Krow[k0], Krow[k0 + 1] };
            acc = wmma_f32(qfrag[kk], bf, acc);
        }
        const float mval = (float)attn_mask[b * SEQ + nb + l16];     // key column
        const float madd = (1.0f - mval) * -10000.0f;
#pragma unroll
        for (int r = 0; r < 8; ++r)
            probs[8 * half + r][nb + l16] = acc[r] * scale + madd;
    }
    __syncthreads();   // single-wave WG: S_NOP barrier + compiler LDS fence

    // ---- row softmax over 512 keys: lane pair (l16, l16+16) shares row l16 ----
    {
        const int row = l16;
        const int c0  = half * (SEQ / 2);
        float mx = -INFINITY;
        for (int c = 0; c < SEQ / 2; ++c) mx = fmaxf(mx, probs[row][c0 + c]);
        mx = fmaxf(mx, __shfl_xor(mx, 16, 32));
        float sum = 0.0f;
        for (int c = 0; c < SEQ / 2; ++c) {
            float e = __expf(probs[row][c0 + c] - mx);
            probs[row][c0 + c] = e;
            sum += e;
        }
        sum += __shfl_xor(sum, 16, 32);
        const float inv = 1.0f / sum;
        for (int c = 0; c < SEQ / 2; ++c) probs[row][c0 + c] *= inv;
    }
    __syncthreads();

    // ---- ctx = probs(16x512) @ V(512x32): two 16x16 n-tiles ----
#pragma unroll
    for (int nt = 0; nt < 2; ++nt) {
        const int nb = nt << 4;
        v8f acc = {};
#pragma unroll 4
        for (int kk = 0; kk < SEQ / 4; ++kk) {
            const int k0 = kk * 4 + 2 * half;
            v2f a  = (v2f){ probs[l16][k0], probs[l16][k0 + 1] };
            v2f bf = (v2f){ V[(size_t)k0 * DIMC + nb + l16],
                            V[(size_t)(k0 + 1) * DIMC + nb + l16] };
            acc = wmma_f32(a, bf, acc);
        }
#pragma unroll
        for (int r = 0; r < 8; ++r) {
            const int q = qbase + 8 * half + r;
            out[((size_t)b * SEQ + q) * DIMC + h * HD + nb + l16] = acc[r];
        }
    }
}

// ---------------------------------------------------------------------------
// Host launcher
// ---------------------------------------------------------------------------
extern "C" void kernel_launch(void* const* d_in, const int* in_sizes, int n_in,
                              void* d_out, int out_size, void* d_ws, size_t ws_size,
                              hipStream_t stream) {
    const float* hidden = (const float*)d_in[0];
    // d_in[1] (time_k) and d_in[2] (time_v) are dead in the reference: never read.
    const int*   mask   = (const int*)  d_in[3];
    const float* Wq     = (const float*)d_in[4];
    const float* bq     = (const float*)d_in[5];
    const float* Wk     = (const float*)d_in[6];
    const float* bk     = (const float*)d_in[7];
    const float* Wv     = (const float*)d_in[8];
    const float* bv     = (const float*)d_in[9];

    float* qkv = (float*)d_ws;   // needs 3 * 1024 * 256 * 4 = 3 MB of scratch

    // 3072 output tiles, 1 wave each -> 384 blocks x 256 threads (8 waves)
    mha_qkv_proj_kernel<<<384, 256, 0, stream>>>(hidden, Wq, bq, Wk, bk, Wv, bv, qkv);

    // 512 (b,h,q-tile) tiles, one single-wave workgroup each
    mha_attn_kernel<<<512, 32, 0, stream>>>(qkv, mask, (float*)d_out);
}